// BatchGFR_84550726189063
// MI455X (gfx1250) — compile-verified
//
#include <hip/hip_runtime.h>
#include <hip/hip_bf16.h>
#include <math.h>

typedef __attribute__((ext_vector_type(2))) float v2f;
typedef __attribute__((ext_vector_type(8))) float v8f;

#define T_DIM 128
#define B_DIM 128
#define N_DIM 256
#define K_DIM 64
#define L_DIM 32
#define ND1   6   // D+1 polynomial coefficients

#define APAD  128           // zero-padded filter: a at [48..111]
#define CROWS (64 + T_DIM)  // 64 leading zero rows + 128 current rows

// ---------------------------------------------------------------------------
// Phase 1: x[t,b,n] = sum_k a[n,k] * cur[t-63+k, b, n]   (zero-padded causal FIR)
// Banded-Toeplitz GEMM per neuron via FP32 WMMA 16x16x4.
// Grid: N * (B/16) blocks of 256 threads (8 waves); wave w owns t-block w.
// All band/boundary handling is done with zero padding in LDS so the
// 5x4 = 20 WMMA MAC loop is branch-free and fully unrolled.
// ---------------------------------------------------------------------------
__global__ __launch_bounds__(256) void fir_wmma_kernel(
    const float* __restrict__ cur,   // [T,B,N]
    const float* __restrict__ a,     // [N,K]
    float* __restrict__ xws)         // [T,B,N]
{
    __shared__ float lds_apad[APAD];        // zero-padded filter
    __shared__ float lds_cp[CROWS * 16];    // [64 zero rows | currents rows] x 16 batch

    const int n  = blockIdx.x >> 3;         // neuron
    const int b0 = (blockIdx.x & 7) * 16;   // batch tile origin
    const int tid = threadIdx.x;

    // padded filter: apad[i] = a[n, i-48] for i in [48,112), else 0
    if (tid < APAD) {
        float v = 0.f;
        if (tid >= 48 && tid < 48 + K_DIM) v = a[n * K_DIM + (tid - 48)];
        lds_apad[tid] = v;
    }
    // zero rows 0..63 (1024 floats)
#pragma unroll
    for (int j = 0; j < 4; ++j) lds_cp[tid + j * 256] = 0.f;
    // currents rows 64..191 (2048 floats)
#pragma unroll
    for (int j = 0; j < 8; ++j) {
        int idx = tid + j * 256;            // 0..2047
        int t   = idx >> 4;
        int bb  = idx & 15;
        lds_cp[64 * 16 + idx] =
            cur[(size_t)t * (B_DIM * N_DIM) + (size_t)(b0 + bb) * N_DIM + n];
    }
    __syncthreads();

    const int wave  = tid >> 5;             // 0..7 -> output t-block
    const int lane  = tid & 31;
    const int m     = lane & 15;            // A row / B col within tile
    const int klane = (lane >> 4) * 2;      // K offset for this lane-half
    const int t0    = wave * 16;

    // A address: apad[111 - (t - s)] ; t = t0+m, s = 16*(wave-4+Joff)+4q+klane
    //   => index = 47 - m + klane + 16*Joff + 4*q   (t0 cancels: shift-invariant)
    const int baseA = 47 - m + klane;
    // B address: padded row = s + 64 = t0 + 16*Joff + 4*q + klane
    const int baseC = (t0 + klane) * 16 + m;

    v8f acc = {0.f, 0.f, 0.f, 0.f, 0.f, 0.f, 0.f, 0.f};

#pragma unroll
    for (int Joff = 0; Joff < 5; ++Joff) {
#pragma unroll
        for (int q = 0; q < 4; ++q) {
            const int ao = baseA + 16 * Joff + 4 * q;
            v2f Af = { lds_apad[ao], lds_apad[ao + 1] };
            const int co = baseC + (16 * Joff + 4 * q) * 16;
            v2f Bf = { lds_cp[co], lds_cp[co + 16] };
            acc = __builtin_amdgcn_wmma_f32_16x16x4_f32(
                false, Af, false, Bf, (short)0, acc, false, false);
        }
    }

    // C/D layout: VGPR r -> (M=r, N=lane) lanes 0-15 ; (M=r+8, N=lane-16) lanes 16-31
    const int hi = lane >> 4;
#pragma unroll
    for (int r = 0; r < 8; ++r) {
        int tt = t0 + hi * 8 + r;
        int bb = b0 + m;
        xws[(size_t)tt * (B_DIM * N_DIM) + (size_t)bb * N_DIM + n] = acc[r];
    }
}

// ---------------------------------------------------------------------------
// Phase 2: sequential feedback scan. One thread per (b, n) channel.
// Block = one batch b (256 threads = all neurons n -> coalesced x/out access).
// Feedback history kept as a register ring buffer (indices constant after
// unrolling in chunks of L=32).
// ---------------------------------------------------------------------------
template <bool USE_WS>
__global__ __launch_bounds__(256) void scan_kernel(
    const float* __restrict__ xws,      // [T,B,N] precomputed x (if USE_WS)
    const float* __restrict__ cur,      // [T,B,N] (fallback path)
    const float* __restrict__ a,        // [N,K]   (fallback path)
    const float* __restrict__ bmat,     // [N,L]
    const float* __restrict__ pc,       // [N,D+1]
    const float* __restrict__ gb_arr,   // [N]
    const float* __restrict__ mc_arr,   // [N]
    const float* __restrict__ mfr_arr,  // [N]
    float* __restrict__ out)            // [T,B,N]
{
    const int n = threadIdx.x;          // 0..255
    const int b = blockIdx.x;           // 0..127

    float bc[L_DIM];
#pragma unroll
    for (int l = 0; l < L_DIM; ++l) bc[l] = bmat[n * L_DIM + l];

    float c2[ND1];
#pragma unroll
    for (int d = 0; d < ND1; ++d) { float c = pc[n * ND1 + d]; c2[d] = c * c; }

    const float gb     = gb_arr[n];
    const float inv_mc = 1.0f / mc_arr[n];
    const float mfr    = mfr_arr[n];

    float ak[K_DIM];
    if (!USE_WS) {
#pragma unroll
        for (int k = 0; k < K_DIM; ++k) ak[k] = a[n * K_DIM + k];
    }

    float fh[L_DIM];                    // ring buffer of last 32 f values
#pragma unroll
    for (int l = 0; l < L_DIM; ++l) fh[l] = 0.f;

    const size_t stride = (size_t)B_DIM * N_DIM;
    const size_t base   = (size_t)b * N_DIM + n;

    for (int tb = 0; tb < T_DIM; tb += L_DIM) {       // tb multiple of 32
#pragma unroll
        for (int u = 0; u < L_DIM; ++u) {             // ring indices constant
            const int t = tb + u;
            // y_t = sum_l b[n,l] * f[t-32+l] ; slot of f[tau] is tau & 31
            float y = 0.f;
#pragma unroll
            for (int l = 0; l < L_DIM; ++l)
                y += bc[l] * fh[(u + l) & (L_DIM - 1)];

            float x;
            if (USE_WS) {
                x = xws[(size_t)t * stride + base];
            } else {
                x = 0.f;
                for (int k = 0; k < K_DIM; ++k) {
                    int ts = t - (K_DIM - 1) + k;
                    if (ts >= 0) x += ak[k] * cur[(size_t)ts * stride + base];
                }
            }

            const float z  = x + 1000.0f * y;
            const float xn = (z - gb) * inv_mc;
            float poly = c2[5];
            poly = poly * xn + c2[4];
            poly = poly * xn + c2[3];
            poly = poly * xn + c2[2];
            poly = poly * xn + c2[1];
            poly = poly * xn + c2[0];
            const float f = fmaxf(0.f, mfr * tanhf(poly));

            fh[u & (L_DIM - 1)] = f;                  // overwrite f[t-32]
            out[(size_t)t * stride + base] = f;
        }
    }
}

// ---------------------------------------------------------------------------
extern "C" void kernel_launch(void* const* d_in, const int* in_sizes, int n_in,
                              void* d_out, int out_size, void* d_ws, size_t ws_size,
                              hipStream_t stream) {
    const float* cur  = (const float*)d_in[0];  // [T,B,N]
    const float* a    = (const float*)d_in[1];  // [N,K]
    const float* bm   = (const float*)d_in[2];  // [N,L]
    const float* pc   = (const float*)d_in[3];  // [N,D+1]
    const float* gb   = (const float*)d_in[4];  // [N]
    const float* mc   = (const float*)d_in[5];  // [N]
    const float* mfr  = (const float*)d_in[6];  // [N]
    float* out = (float*)d_out;

    const size_t need = (size_t)T_DIM * B_DIM * N_DIM * sizeof(float);
    if (d_ws != nullptr && ws_size >= need) {
        float* xws = (float*)d_ws;
        fir_wmma_kernel<<<dim3(N_DIM * (B_DIM / 16)), dim3(256), 0, stream>>>(cur, a, xws);
        scan_kernel<true><<<dim3(B_DIM), dim3(256), 0, stream>>>(
            xws, cur, a, bm, pc, gb, mc, mfr, out);
    } else {
        scan_kernel<false><<<dim3(B_DIM), dim3(256), 0, stream>>>(
            nullptr, cur, a, bm, pc, gb, mc, mfr, out);
    }
}